// MAGNet_82197084110904
// MI455X (gfx1250) — compile-verified
//
#include <hip/hip_runtime.h>

// ---------------- problem constants (from the reference) ----------------
#define NN      50000
#define DD      64
#define EE      800000
#define LL      3
#define NITERS  14
#define NTILES  (NN / 16)          // 3125, exact
#define WAVES_PER_BLOCK 4
#define SPMM_BLOCK (WAVES_PER_BLOCK * 32)

typedef float v2f __attribute__((ext_vector_type(2)));
typedef float v8f __attribute__((ext_vector_type(8)));

__device__ __forceinline__ int rfl_i(int x) {
  return __builtin_amdgcn_readfirstlane(x);
}
__device__ __forceinline__ float rfl_f(float x) {
  return __builtin_bit_cast(float, __builtin_amdgcn_readfirstlane(__builtin_bit_cast(int, x)));
}
__device__ __forceinline__ float softth(float x, float t) {
  return fmaxf(x - t, 0.0f) - fmaxf(-x - t, 0.0f);
}

// ---------------- setup: num_base = d*feat*mask^2 ; denom = d*mask^2 + 1 ----------------
__global__ void k_setup(const float* __restrict__ feat, const float* __restrict__ dv,
                        const float* __restrict__ mask, float* __restrict__ numb,
                        float* __restrict__ den) {
  int i = blockIdx.x * blockDim.x + threadIdx.x;
  if (i >= NN * DD) return;
  int row = i >> 6;
  float m2 = mask[row] * mask[row];
  float d  = dv[row];
  numb[i] = d * feat[i] * m2;
  if ((i & 63) == 0) den[row] = d * m2 + 1.0f;   // GAMMA = 1
}

// ---------------- CSR build ----------------
__global__ void k_count(const int* __restrict__ wr, int* __restrict__ cnt) {
  int e = blockIdx.x * blockDim.x + threadIdx.x;
  if (e >= EE) return;
  atomicAdd(&cnt[wr[e]], 1);
}

__global__ void k_scan1(const int* __restrict__ cnt, int* __restrict__ rp,
                        int* __restrict__ bsum) {
  __shared__ int s[256];
  int i = blockIdx.x * 256 + threadIdx.x;
  s[threadIdx.x] = (i < NN) ? cnt[i] : 0;
  __syncthreads();
  for (int off = 1; off < 256; off <<= 1) {
    int t = (threadIdx.x >= off) ? s[threadIdx.x - off] : 0;
    __syncthreads();
    s[threadIdx.x] += t;
    __syncthreads();
  }
  if (i < NN) rp[i + 1] = s[threadIdx.x];
  if (threadIdx.x == 255) bsum[blockIdx.x] = s[255];
}

__global__ void k_scan2(int* __restrict__ bsum, int* __restrict__ rp, int nb) {
  __shared__ int s[256];
  int t = threadIdx.x;
  s[t] = (t < nb) ? bsum[t] : 0;
  __syncthreads();
  for (int off = 1; off < 256; off <<= 1) {
    int x = (t >= off) ? s[t - off] : 0;
    __syncthreads();
    s[t] += x;
    __syncthreads();
  }
  if (t < nb) bsum[t] = s[t];
  if (t == 0) rp[0] = 0;
}

__global__ void k_scan3(int* __restrict__ rp, const int* __restrict__ bsum) {
  int i = blockIdx.x * blockDim.x + threadIdx.x;
  if (i >= NN) return;
  int b = i >> 8;
  if (b > 0) rp[i + 1] += bsum[b - 1];
}

__global__ void k_cursor_init(const int* __restrict__ rp, int* __restrict__ cur) {
  int i = blockIdx.x * blockDim.x + threadIdx.x;
  if (i >= NN) return;
  cur[i] = rp[i];
}

__global__ void k_scatter(const int* __restrict__ wr, const int* __restrict__ wc,
                          const float* __restrict__ wv, int* __restrict__ cur,
                          int* __restrict__ col_s, int* __restrict__ lrow_s,
                          float* __restrict__ vals_s) {
  int e = blockIdx.x * blockDim.x + threadIdx.x;
  if (e >= EE) return;
  int r = wr[e];
  int p = atomicAdd(&cur[r], 1);
  col_s[p]  = wc[e];
  lrow_s[p] = r & 15;
#pragma unroll
  for (int l = 0; l < LL; ++l) vals_s[(size_t)l * EE + p] = wv[(size_t)l * EE + e];
}

// ---------------- fused SpMM kernels (one wave32 per 16-row tile) ----------------
// WMMA 16x16x4 F32: A = 16 rows x 4 edges (per-layer vals, masked by local row),
//   lanes 0-15 hold M=0..15 {K0,K1}, lanes 16-31 hold M=0..15 {K2,K3}.
// B = 4 edges x 16 cols (gathered feature rows), mirrored layout:
//   lanes 0-15 hold N=0..15 {K0,K1}, lanes 16-31 hold N=0..15 {K2,K3}.
// C/D vgpr j: lanes 0-15 -> M=j, lanes 16-31 -> M=j+8, N = lane&15.
//
// Tail handling is fully branchless: out-of-range chunk slots clamp their edge
// index to p1-1 (valid once the loop is entered); loads always execute from the
// clamped address and lrow is s_cselect'ed to -1 afterwards, which structurally
// zeroes the A-matrix entry and annihilates the garbage gather.

// Kernel A: wu[l] = W_l @ U ; y = (first ? 0 : v + wu) ; v' = y - soft(y + wu, nu_l * d)
__global__ __launch_bounds__(SPMM_BLOCK) void k_spmm_A(
    const float* __restrict__ Xg,        // U_k  (N x 64)
    float* __restrict__ v,               // L x N x 64 (in/out)
    const float* __restrict__ dvec,
    const int* __restrict__ rp,
    const int* __restrict__ col_s,
    const int* __restrict__ lrow_s,
    const float* __restrict__ vals_s,
    int first) {
  int tile = blockIdx.x * WAVES_PER_BLOCK + (threadIdx.x >> 5);
  if (tile >= NTILES) return;                // wave-uniform exit: EXEC stays all-ones
  int lane = threadIdx.x & 31;
  int r0 = tile << 4;
  int p0 = rfl_i(rp[r0]);
  int p1 = rfl_i(rp[r0 + 16]);
  int kh   = (lane < 16) ? 0 : 2;
  int nIdx = lane & 15;

  v8f acc[LL][4];
#pragma unroll
  for (int l = 0; l < LL; ++l)
#pragma unroll
    for (int dc = 0; dc < 4; ++dc) acc[l][dc] = (v8f)0.0f;

  for (int p = p0; p < p1; p += 4) {
    int col[4], lr[4];
    float val[LL][4];
#pragma unroll
    for (int k = 0; k < 4; ++k) {
      int q     = p + k;
      bool vld  = q < p1;                 // wave-uniform -> s_cselect, no branch
      int qc    = vld ? q : (p1 - 1);
      col[k]    = rfl_i(col_s[qc]);
      int lrv   = rfl_i(lrow_s[qc]);      // always load (address valid), ...
      lr[k]     = vld ? lrv : -1;         // ... then select: keeps loads clause-able
#pragma unroll
      for (int l = 0; l < LL; ++l) val[l][k] = rfl_f(vals_s[(size_t)l * EE + qc]);
    }
    // prefetch next chunk's gathered rows + metadata (CDNA5 global_prefetch_b8)
    if (p + 4 < p1) {
      int cn = col_s[p + 4];
      __builtin_prefetch(&Xg[(size_t)cn * DD], 0, 0);
      __builtin_prefetch(&Xg[(size_t)cn * DD + 32], 0, 0);
      __builtin_prefetch(&col_s[p + 8], 0, 0);
      __builtin_prefetch(&vals_s[p + 8], 0, 0);
    }

    v2f A[LL];
#pragma unroll
    for (int l = 0; l < LL; ++l) {
      A[l].x = (lr[kh]     == nIdx) ? val[l][kh]     : 0.0f;
      A[l].y = (lr[kh + 1] == nIdx) ? val[l][kh + 1] : 0.0f;
    }
    const float* ra = &Xg[(size_t)col[kh] * DD];
    const float* rb = &Xg[(size_t)col[kh + 1] * DD];
#pragma unroll
    for (int dc = 0; dc < 4; ++dc) {
      v2f B;
      B.x = ra[dc * 16 + nIdx];
      B.y = rb[dc * 16 + nIdx];
#pragma unroll
      for (int l = 0; l < LL; ++l)
        acc[l][dc] = __builtin_amdgcn_wmma_f32_16x16x4_f32(
            false, A[l], false, B, (short)0, acc[l][dc], false, false);
    }
  }

  // epilogue: soft-threshold update of v
  const float nu[LL] = {0.0f, 1.0f, 0.25f};
  int mBase = (lane < 16) ? 0 : 8;
#pragma unroll
  for (int j = 0; j < 8; ++j) {
    int row = r0 + j + mBase;
    float d = dvec[row];
#pragma unroll
    for (int l = 0; l < LL; ++l) {
      float eta = nu[l] * d;
#pragma unroll
      for (int dc = 0; dc < 4; ++dc) {
        size_t idx = ((size_t)l * NN + row) * DD + dc * 16 + nIdx;
        float wu = acc[l][dc][j];
        float y  = first ? 0.0f : (v[idx] + wu);
        float x  = y + wu;
        v[idx] = y - softth(x, eta);
      }
    }
  }
}

// Kernel B: wtv = sum_l W_l @ v[l] ; U_out = (num_base - wtv) / denom
__global__ __launch_bounds__(SPMM_BLOCK) void k_spmm_B(
    const float* __restrict__ v,         // L x N x 64
    float* __restrict__ Uout,            // N x 64
    const float* __restrict__ numb,
    const float* __restrict__ den,
    const int* __restrict__ rp,
    const int* __restrict__ col_s,
    const int* __restrict__ lrow_s,
    const float* __restrict__ vals_s) {
  int tile = blockIdx.x * WAVES_PER_BLOCK + (threadIdx.x >> 5);
  if (tile >= NTILES) return;
  int lane = threadIdx.x & 31;
  int r0 = tile << 4;
  int p0 = rfl_i(rp[r0]);
  int p1 = rfl_i(rp[r0 + 16]);
  int kh   = (lane < 16) ? 0 : 2;
  int nIdx = lane & 15;

  v8f acc[4];
#pragma unroll
  for (int dc = 0; dc < 4; ++dc) acc[dc] = (v8f)0.0f;

  for (int p = p0; p < p1; p += 4) {
    int col[4], lr[4];
    float val[LL][4];
#pragma unroll
    for (int k = 0; k < 4; ++k) {
      int q     = p + k;
      bool vld  = q < p1;
      int qc    = vld ? q : (p1 - 1);
      col[k]    = rfl_i(col_s[qc]);
      int lrv   = rfl_i(lrow_s[qc]);
      lr[k]     = vld ? lrv : -1;
#pragma unroll
      for (int l = 0; l < LL; ++l) val[l][k] = rfl_f(vals_s[(size_t)l * EE + qc]);
    }
    if (p + 4 < p1) {
      int cn = col_s[p + 4];
#pragma unroll
      for (int l = 0; l < LL; ++l) {
        __builtin_prefetch(&v[((size_t)l * NN + cn) * DD], 0, 0);
        __builtin_prefetch(&v[((size_t)l * NN + cn) * DD + 32], 0, 0);
      }
      __builtin_prefetch(&col_s[p + 8], 0, 0);
      __builtin_prefetch(&vals_s[p + 8], 0, 0);
    }

#pragma unroll
    for (int l = 0; l < LL; ++l) {
      v2f A;
      A.x = (lr[kh]     == nIdx) ? val[l][kh]     : 0.0f;
      A.y = (lr[kh + 1] == nIdx) ? val[l][kh + 1] : 0.0f;
      const float* ra = &v[((size_t)l * NN + col[kh]) * DD];
      const float* rb = &v[((size_t)l * NN + col[kh + 1]) * DD];
#pragma unroll
      for (int dc = 0; dc < 4; ++dc) {
        v2f B;
        B.x = ra[dc * 16 + nIdx];
        B.y = rb[dc * 16 + nIdx];
        acc[dc] = __builtin_amdgcn_wmma_f32_16x16x4_f32(
            false, A, false, B, (short)0, acc[dc], false, false);
      }
    }
  }

  int mBase = (lane < 16) ? 0 : 8;
#pragma unroll
  for (int j = 0; j < 8; ++j) {
    int row = r0 + j + mBase;
    float dn = den[row];
#pragma unroll
    for (int dc = 0; dc < 4; ++dc) {
      size_t idx = (size_t)row * DD + dc * 16 + nIdx;
      Uout[idx] = (numb[idx] - acc[dc][j]) / dn;
    }
  }
}

// ---------------- host side ----------------
extern "C" void kernel_launch(void* const* d_in, const int* in_sizes, int n_in,
                              void* d_out, int out_size, void* d_ws, size_t ws_size,
                              hipStream_t stream) {
  const float* feat = (const float*)d_in[0];
  const int*   wr   = (const int*)d_in[1];
  const int*   wc   = (const int*)d_in[2];
  const float* wv   = (const float*)d_in[3];
  const float* dv   = (const float*)d_in[4];
  const float* mask = (const float*)d_in[5];
  float* out = (float*)d_out;

  // carve workspace (all 4-byte elements)
  float* vbuf   = (float*)d_ws;                  // L*N*64
  float* Ubuf   = vbuf + (size_t)LL * NN * DD;   // N*64
  float* numb   = Ubuf + (size_t)NN * DD;        // N*64
  float* den    = numb + (size_t)NN * DD;        // N
  int*   rp     = (int*)(den + NN);              // N+1
  int*   cursor = rp + (NN + 1);                 // N
  int*   bsum   = cursor + NN;                   // 256
  int*   col_s  = bsum + 256;                    // E
  int*   lrow_s = col_s + EE;                    // E
  float* vals_s = (float*)(lrow_s + EE);         // L*E

  const int TPB = 256;
  int nbScan = (NN + 255) / 256;                 // 196

  hipMemsetAsync(cursor, 0, (size_t)NN * sizeof(int), stream);
  k_setup<<<(NN * DD + TPB - 1) / TPB, TPB, 0, stream>>>(feat, dv, mask, numb, den);
  k_count<<<(EE + TPB - 1) / TPB, TPB, 0, stream>>>(wr, cursor);
  k_scan1<<<nbScan, 256, 0, stream>>>(cursor, rp, bsum);
  k_scan2<<<1, 256, 0, stream>>>(bsum, rp, nbScan);
  k_scan3<<<(NN + TPB - 1) / TPB, TPB, 0, stream>>>(rp, bsum);
  k_cursor_init<<<(NN + TPB - 1) / TPB, TPB, 0, stream>>>(rp, cursor);
  k_scatter<<<(EE + TPB - 1) / TPB, TPB, 0, stream>>>(wr, wc, wv, cursor, col_s, lrow_s, vals_s);

  int spmmGrid = (NTILES + WAVES_PER_BLOCK - 1) / WAVES_PER_BLOCK;
  for (int it = 0; it < NITERS; ++it) {
    const float* Uin = (it == 0) ? feat : Ubuf;
    k_spmm_A<<<spmmGrid, SPMM_BLOCK, 0, stream>>>(Uin, vbuf, dv, rp, col_s, lrow_s,
                                                  vals_s, it == 0 ? 1 : 0);
    float* Uout = (it == NITERS - 1) ? out : Ubuf;
    k_spmm_B<<<spmmGrid, SPMM_BLOCK, 0, stream>>>(vbuf, Uout, numb, den, rp, col_s,
                                                  lrow_s, vals_s);
  }
}